// FRAP_47047071761018
// MI455X (gfx1250) — compile-verified
//
#include <hip/hip_runtime.h>

// FRAP forward pass for MI455X (gfx1250): single workgroup, 8 wave32s,
// all intermediates in LDS, every GEMM on V_WMMA_F32_16X16X4_F32 (full f32).

#define LDA 260   // LDS row stride in floats; 260 % 64 == 4 -> conflict-free frag loads

typedef __attribute__((ext_vector_type(2))) float v2f;
typedef __attribute__((ext_vector_type(8))) float v8f;

struct FrapParams {
  const float *waiting, *phase, *mu, *sigma, *wmax, *ipos, *jpos;
  const float *dw1, *db1, *dw2, *db2, *dw3, *db3;
  const float *cw1, *cb1, *cw2, *cb2, *cw3, *cb3;
  const float *Dw1, *Db1, *Dw2, *Db2, *Dw3, *Db3;
  const float *Cw1, *Cb1, *Cw2, *Cb2, *Cw3, *Cb3, *Cw4, *Cb4;
  float* out;
};

// O(Mrows x N) = act( A(Mrows x K) * W(N x K)^T + bias ), tiles padded to 16.
// A in LDS (zero-padded to Mtiles*16 x roundup4(K)); W,bias in global.
// Writes the full padded tile region of Osh, storing 0 outside (Mrows, N) so
// Osh is a valid zero-padded A-matrix for the next layer.
__device__ __forceinline__ void gemm_wmma_f32(
    const float* __restrict__ Ash, const float* __restrict__ W,
    const float* __restrict__ bias, float* __restrict__ Osh,
    int Mtiles, int Mrows, int N, int K, bool relu)
{
  const int lane = threadIdx.x & 31;
  const int wave = threadIdx.x >> 5;
  const int r    = lane & 15;   // row-in-tile for A/B fragments, col for C/D
  const int kh   = lane >> 4;   // K-half select per ISA 16x4 / 4x16 layouts
  const int Ntiles = (N + 15) >> 4;
  const int Ksteps = (K + 3) >> 2;
  const int total  = Mtiles * Ntiles;

  for (int t = wave; t < total; t += 8) {      // wave-uniform: EXEC all-ones at WMMA
    const int mt = t / Ntiles;
    const int nt = t - mt * Ntiles;
    v8f acc = {};
    const float* Arow = Ash + (mt * 16 + r) * LDA;
    const int  wrow = nt * 16 + r;
    const bool wv   = wrow < N;
    const float* Wrow = W + (wv ? wrow : 0) * K;

    for (int ks = 0; ks < Ksteps; ++ks) {
      const int k0 = ks * 4 + 2 * kh;          // frag elem v holds k = k0 + v
      v2f a = *(const v2f*)(Arow + k0);        // LDS zero-padded: always safe
      v2f b;
      b.x = (wv && (k0     < K)) ? Wrow[k0]     : 0.f;
      b.y = (wv && (k0 + 1 < K)) ? Wrow[k0 + 1] : 0.f;
      acc = __builtin_amdgcn_wmma_f32_16x16x4_f32(
          /*neg_a=*/false, a, /*neg_b=*/false, b,
          /*c_mod=*/(short)0, acc, /*reuse_a=*/false, /*reuse_b=*/false);
    }

    const int   n  = nt * 16 + r;
    const float bn = (n < N) ? bias[n] : 0.f;
#pragma unroll
    for (int v = 0; v < 8; ++v) {
      const int m = mt * 16 + v + 8 * kh;      // C/D layout: row = v + 8*(lane>=16)
      float val = acc[v] + bn;
      if (relu) val = fmaxf(val, 0.f);
      Osh[m * LDA + n] = (m < Mrows && n < N) ? val : 0.f;
    }
  }
}

__global__ __launch_bounds__(256) void frap_forward_kernel(FrapParams p) {
  __shared__ __align__(16) float bufA[64 * LDA];
  __shared__ __align__(16) float bufB[64 * LDA];
  const int tid = threadIdx.x;

  // ---- feat (8 x 14) -> bufA, zero-padded 16x16 ----
  {
    const int r = tid >> 4, c = tid & 15;
    float v = 0.f;
    if (r < 8 && c < 14) {
      switch (c) {
        case 0:  v = p.ipos[0];      break;
        case 1:  v = p.jpos[0];      break;
        case 2:  v = p.mu[r];        break;
        case 3:  v = p.sigma[r];     break;
        case 4:  v = p.wmax[r];      break;
        case 5:  v = p.waiting[r];   break;
        default: v = p.phase[c - 6]; break;
      }
    }
    bufA[r * LDA + c] = v;
  }
  __syncthreads();

  // ---- NN_demand: 14 -> 256 -> 32 -> 4 ----
  gemm_wmma_f32(bufA, p.dw1, p.db1, bufB, 1, 8, 256, 14,  true);  __syncthreads();
  gemm_wmma_f32(bufB, p.dw2, p.db2, bufA, 1, 8, 32,  256, true);  __syncthreads();
  gemm_wmma_f32(bufA, p.dw3, p.db3, bufB, 1, 8, 4,   32,  false); __syncthreads();
  // demand (8x4) now in bufB

  // ---- vec (8 x 8) = [demand[pairs[:,0]], demand[pairs[:,1]]] -> bufA ----
  // PAIRS rows: PA(p)=p&3 ; PB(p)= p<4 ? p+4 : ((p&3)^1)+4
  {
    const int r = tid >> 4, c = tid & 15;
    float v = 0.f;
    if (r < 8 && c < 8) {
      const int src = (c < 4) ? (r & 3)
                              : ((r < 4) ? (r + 4) : (((r & 3) ^ 1) + 4));
      v = bufB[src * LDA + (c & 3)];
    }
    bufA[r * LDA + c] = v;
  }
  __syncthreads();

  // ---- NN_coupled_demand: 8 -> 128 -> 32 -> 4 ----
  gemm_wmma_f32(bufA, p.cw1, p.cb1, bufB, 1, 8, 128, 8,   true);  __syncthreads();
  gemm_wmma_f32(bufB, p.cw2, p.cb2, bufA, 1, 8, 32,  128, true);  __syncthreads();
  gemm_wmma_f32(bufA, p.cw3, p.cb3, bufB, 1, 8, 4,   32,  false); __syncthreads();
  // dp (8x4) now in bufB

  // ---- D tensor (8ch, 8, 7) as 56 positions x 8 channels -> bufA (64 x 8) ----
  for (int idx = tid; idx < 512; idx += 256) {
    const int rr = idx >> 3, c = idx & 7;
    float v = 0.f;
    if (rr < 56) {
      const int i = rr / 7, k = rr - 7 * i;
      if (c < 4) v = bufB[i * LDA + c];                    // D[c,i,k]   = dp[i][c]
      else { const int j = k + (k >= i ? 1 : 0);
             v = bufB[j * LDA + (c - 4)]; }                // D[4+c,i,k] = dp[j][c]
    }
    bufA[rr * LDA + c] = v;
  }
  __syncthreads();

  // ---- Conv_D (1x1 = channel GEMMs over 56 positions): 8 -> 256 -> 32 -> 2 ----
  gemm_wmma_f32(bufA, p.Dw1, p.Db1, bufB, 4, 56, 256, 8,   true);  __syncthreads();
  gemm_wmma_f32(bufB, p.Dw2, p.Db2, bufA, 4, 56, 32,  256, true);  __syncthreads();
  gemm_wmma_f32(bufA, p.Dw3, p.Db3, bufB, 4, 56, 2,   32,  false); __syncthreads();
  // H_c = [H_d ; zeros] is implicit: bufB cols 2..15 were stored as 0.

  // ---- Conv_C: 3 -> 256 -> 32 -> 8 (all relu) ----
  gemm_wmma_f32(bufB, p.Cw1, p.Cb1, bufA, 4, 56, 256, 3,   true);  __syncthreads();
  gemm_wmma_f32(bufA, p.Cw2, p.Cb2, bufB, 4, 56, 32,  256, true);  __syncthreads();
  gemm_wmma_f32(bufB, p.Cw3, p.Cb3, bufA, 4, 56, 8,   32,  true);  __syncthreads();

  // ---- final (1,7) conv: out[h] = sum_{c,w} Cw4[c][w] * hmap[c][h][w] + b ----
  if (tid < 8) {
    float s = p.Cb4[0];
#pragma unroll
    for (int c = 0; c < 8; ++c)
#pragma unroll
      for (int w = 0; w < 7; ++w)
        s += p.Cw4[c * 7 + w] * bufA[(tid * 7 + w) * LDA + c];
    p.out[tid] = s;
  }
}

extern "C" void kernel_launch(void* const* d_in, const int* in_sizes, int n_in,
                              void* d_out, int out_size, void* d_ws, size_t ws_size,
                              hipStream_t stream) {
  (void)in_sizes; (void)n_in; (void)out_size; (void)d_ws; (void)ws_size;
  FrapParams p;
  p.waiting = (const float*)d_in[0];
  p.phase   = (const float*)d_in[1];
  p.mu      = (const float*)d_in[2];
  p.sigma   = (const float*)d_in[3];
  p.wmax    = (const float*)d_in[4];
  p.ipos    = (const float*)d_in[5];
  p.jpos    = (const float*)d_in[6];
  p.dw1 = (const float*)d_in[7];  p.db1 = (const float*)d_in[8];
  p.dw2 = (const float*)d_in[9];  p.db2 = (const float*)d_in[10];
  p.dw3 = (const float*)d_in[11]; p.db3 = (const float*)d_in[12];
  p.cw1 = (const float*)d_in[13]; p.cb1 = (const float*)d_in[14];
  p.cw2 = (const float*)d_in[15]; p.cb2 = (const float*)d_in[16];
  p.cw3 = (const float*)d_in[17]; p.cb3 = (const float*)d_in[18];
  p.Dw1 = (const float*)d_in[19]; p.Db1 = (const float*)d_in[20];
  p.Dw2 = (const float*)d_in[21]; p.Db2 = (const float*)d_in[22];
  p.Dw3 = (const float*)d_in[23]; p.Db3 = (const float*)d_in[24];
  p.Cw1 = (const float*)d_in[25]; p.Cb1 = (const float*)d_in[26];
  p.Cw2 = (const float*)d_in[27]; p.Cb2 = (const float*)d_in[28];
  p.Cw3 = (const float*)d_in[29]; p.Cb3 = (const float*)d_in[30];
  p.Cw4 = (const float*)d_in[31]; p.Cb4 = (const float*)d_in[32];
  p.out = (float*)d_out;
  frap_forward_kernel<<<1, 256, 0, stream>>>(p);
}